// PhysicsLSGStep_54004918780394
// MI455X (gfx1250) — compile-verified
//
#include <hip/hip_runtime.h>
#include <hip/hip_bf16.h>

#define N_NODES 200000
#define N_EDGES 3200000
#define N_FEAT  8
#define NF      (N_NODES * N_FEAT)   // 1,600,000 (== 6250 * 256 exactly)
#define CG_ITERS 30
#define CG_TOL   1e-4f
#define DT_MIN   0.02f
#define DT_MAX   2.0f

#define TPB      256
#define NB_NF    (NF / TPB)          // 6250
#define NB_E     (N_EDGES / TPB)     // 12500
#define NB_DOT   1024

typedef __attribute__((ext_vector_type(2))) float v2f;
typedef __attribute__((ext_vector_type(8))) float v8f;

__device__ __forceinline__ float clamp_dt(const float* dtp) {
  return fminf(fmaxf(dtp[0], DT_MIN), DT_MAX);
}

// scal layout: [0]=rs [1]=pAp_acc [2]=rs_acc [3]=done(gate) [4]=alpha [5]=beta [6]=done_next

// ---------------- setup kernels ----------------

__global__ __launch_bounds__(TPB) void kzero_setup(float* __restrict__ tmp,
                                                   float* __restrict__ x,
                                                   float* __restrict__ scal) {
  int i = blockIdx.x * TPB + threadIdx.x;
  if (i < NF) { tmp[i] = 0.0f; x[i] = 0.0f; }
  if (i < 16) scal[i] = 0.0f;
}

// per-edge precompute: inv_dx, and scatter slope_e to slope_n (stored in tmp[0..N))
__global__ __launch_bounds__(TPB) void kprep(const float* __restrict__ eattr,
                                             const int* __restrict__ dst,
                                             float* __restrict__ inv,
                                             float* __restrict__ slope_n) {
  int e = blockIdx.x * TPB + threadIdx.x;
  if (e >= N_EDGES) return;
  float dx = fmaxf(eattr[2 * e + 0], 1e-6f);
  float iv = 1.0f / dx;
  inv[e] = iv;
  float sl = eattr[2 * e + 1] * iv;
  atomicAdd(&slope_n[dst[e]], sl);
}

// b = u - dt*g*slope_n  (into w); then re-zero the slope region of tmp
__global__ __launch_bounds__(TPB) void kb(const float* __restrict__ u,
                                          float* __restrict__ slope_n,
                                          const float* __restrict__ dtp,
                                          const float* __restrict__ gp,
                                          float* __restrict__ w) {
  int i = blockIdx.x * TPB + threadIdx.x;
  bool ok = (i < NF);
  int n = i >> 3;
  if (ok) {
    float dt = clamp_dt(dtp);
    w[i] = u[i] - dt * gp[0] * slope_n[n];
  }
  __syncthreads();                 // all 8 readers of slope_n[n] are in this block
  if (ok && (i & 7) == 0) slope_n[n] = 0.0f;
}

// r = w + dt*tmp ; p = r
__global__ __launch_bounds__(TPB) void krhs(const float* __restrict__ w,
                                            const float* __restrict__ tmp,
                                            const float* __restrict__ dtp,
                                            float* __restrict__ r,
                                            float* __restrict__ p) {
  int i = blockIdx.x * TPB + threadIdx.x;
  if (i >= NF) return;
  float v = fmaf(clamp_dt(dtp), tmp[i], w[i]);
  r[i] = v;
  p[i] = v;
}

__global__ void kinit_scal(float* __restrict__ scal) {
  scal[0] = scal[2];   // rs = ||r||^2
  scal[1] = 0.0f;
  scal[2] = 0.0f;
  scal[3] = 0.0f;      // done = false
  scal[6] = 0.0f;      // done_next = false
}

// ---------------- per-iteration kernels ----------------

// tmp = 0 ; advance gate (done = done_next) ; zero pAp accumulator
__global__ __launch_bounds__(TPB) void kzero_iter(float* __restrict__ tmp,
                                                  float* __restrict__ scal) {
  int i = blockIdx.x * TPB + threadIdx.x;
  if (i < NF) tmp[i] = 0.0f;
  if (i == 0) { scal[1] = 0.0f; scal[3] = scal[6]; }
}

// D1 scatter: tmp[dst] += (p[dst]-p[src])*inv
__global__ __launch_bounds__(TPB) void kd1(const int* __restrict__ src,
                                           const int* __restrict__ dst,
                                           const float* __restrict__ inv,
                                           const float* __restrict__ p,
                                           float* __restrict__ tmp,
                                           const float* __restrict__ gate) {
  if (gate[0] != 0.0f) return;
  int e = blockIdx.x * TPB + threadIdx.x;
  if (e >= N_EDGES) return;
  int s = src[e], d = dst[e];
  float iv = inv[e];
  const float4* p4 = (const float4*)p;
  float4 pd0 = p4[d * 2 + 0], pd1 = p4[d * 2 + 1];
  float4 ps0 = p4[s * 2 + 0], ps1 = p4[s * 2 + 1];
  float* td = tmp + (size_t)d * 8;
  atomicAdd(td + 0, (pd0.x - ps0.x) * iv);
  atomicAdd(td + 1, (pd0.y - ps0.y) * iv);
  atomicAdd(td + 2, (pd0.z - ps0.z) * iv);
  atomicAdd(td + 3, (pd0.w - ps0.w) * iv);
  atomicAdd(td + 4, (pd1.x - ps1.x) * iv);
  atomicAdd(td + 5, (pd1.y - ps1.y) * iv);
  atomicAdd(td + 6, (pd1.z - ps1.z) * iv);
  atomicAdd(td + 7, (pd1.w - ps1.w) * iv);
}

// w = p + dt*u*tmp ; tmp = 0 (each thread owns its element -> safe fused clear)
__global__ __launch_bounds__(TPB) void kw(const float* __restrict__ p,
                                          const float* __restrict__ u,
                                          float* __restrict__ tmp,
                                          const float* __restrict__ dtp,
                                          float* __restrict__ w,
                                          const float* __restrict__ gate) {
  if (gate[0] != 0.0f) return;
  int i = blockIdx.x * TPB + threadIdx.x;
  if (i >= NF) return;
  w[i] = fmaf(clamp_dt(dtp) * u[i], tmp[i], p[i]);
  tmp[i] = 0.0f;
}

// D1^T scatter of (u*w): yd = u[dst]*w[dst]*inv ; tmp[dst]+=yd ; tmp[src]-=yd
__global__ __launch_bounds__(TPB) void kd1t(const int* __restrict__ src,
                                            const int* __restrict__ dst,
                                            const float* __restrict__ inv,
                                            const float* __restrict__ u,
                                            const float* __restrict__ w,
                                            float* __restrict__ tmp,
                                            const float* __restrict__ gate) {
  if (gate[0] != 0.0f) return;
  int e = blockIdx.x * TPB + threadIdx.x;
  if (e >= N_EDGES) return;
  int s = src[e], d = dst[e];
  float iv = inv[e];
  const float4* u4 = (const float4*)u;
  const float4* w4 = (const float4*)w;
  float4 ud0 = u4[d * 2 + 0], ud1 = u4[d * 2 + 1];
  float4 wd0 = w4[d * 2 + 0], wd1 = w4[d * 2 + 1];
  float y0 = ud0.x * wd0.x * iv, y1 = ud0.y * wd0.y * iv;
  float y2 = ud0.z * wd0.z * iv, y3 = ud0.w * wd0.w * iv;
  float y4 = ud1.x * wd1.x * iv, y5 = ud1.y * wd1.y * iv;
  float y6 = ud1.z * wd1.z * iv, y7 = ud1.w * wd1.w * iv;
  float* td = tmp + (size_t)d * 8;
  float* ts = tmp + (size_t)s * 8;
  atomicAdd(td + 0,  y0); atomicAdd(td + 1,  y1);
  atomicAdd(td + 2,  y2); atomicAdd(td + 3,  y3);
  atomicAdd(td + 4,  y4); atomicAdd(td + 5,  y5);
  atomicAdd(td + 6,  y6); atomicAdd(td + 7,  y7);
  atomicAdd(ts + 0, -y0); atomicAdd(ts + 1, -y1);
  atomicAdd(ts + 2, -y2); atomicAdd(ts + 3, -y3);
  atomicAdd(ts + 4, -y4); atomicAdd(ts + 5, -y5);
  atomicAdd(ts + 6, -y6); atomicAdd(ts + 7, -y7);
}

// Ap (in place over w): w = w + dt*tmp
__global__ __launch_bounds__(TPB) void kap(float* __restrict__ w,
                                           const float* __restrict__ tmp,
                                           const float* __restrict__ dtp,
                                           const float* __restrict__ gate) {
  if (gate[0] != 0.0f) return;
  int i = blockIdx.x * TPB + threadIdx.x;
  if (i >= NF) return;
  w[i] = fmaf(clamp_dt(dtp), tmp[i], w[i]);
}

// dot(a,b) accumulated into *acc. Wave-level reduction via V_WMMA_F32_16X16X4_F32:
// A(16x4) carries one partial per lane (K=1,3 columns zero), B = all-ones, so
// D[m,j] = s(m)+s(m+16); row-summing 8 accum VGPRs + shfl_xor(16) yields the
// full wave32 sum in one matrix op, f32 end-to-end.
__global__ __launch_bounds__(TPB) void kdot(const float* __restrict__ a,
                                            const float* __restrict__ b,
                                            float* __restrict__ acc,
                                            const float* __restrict__ gate,
                                            int n) {
  if (gate[0] != 0.0f) return;     // uniform: whole wave exits, EXEC stays full
  __shared__ float wpart[TPB / 32];
  int tid = blockIdx.x * TPB + threadIdx.x;
  int stride = gridDim.x * TPB;
  float s = 0.0f;
  for (int i = tid; i < n; i += stride) s = fmaf(a[i], b[i], s);
  float total;
#if defined(__gfx1250__) && __has_builtin(__builtin_amdgcn_wmma_f32_16x16x4_f32)
  v2f av; av[0] = s;    av[1] = 0.0f;
  v2f bv; bv[0] = 1.0f; bv[1] = 1.0f;
  v8f c = {0.f, 0.f, 0.f, 0.f, 0.f, 0.f, 0.f, 0.f};
  c = __builtin_amdgcn_wmma_f32_16x16x4_f32(false, av, false, bv,
                                            (short)0, c, false, false);
  float rowsum = c[0] + c[1] + c[2] + c[3] + c[4] + c[5] + c[6] + c[7];
  total = rowsum + __shfl_xor(rowsum, 16, 32);
#else
  total = s;
  for (int off = 16; off > 0; off >>= 1) total += __shfl_xor(total, off, 32);
#endif
  int lane = threadIdx.x & 31;
  int wid = threadIdx.x >> 5;
  if (lane == 0) wpart[wid] = total;
  __syncthreads();
  if (threadIdx.x == 0) {
    float t = 0.0f;
    for (int k = 0; k < TPB / 32; ++k) t += wpart[k];
    atomicAdd(acc, t);
  }
}

__global__ void kalpha(float* __restrict__ scal) {
  if (scal[3] != 0.0f) return;
  scal[4] = scal[0] / (scal[1] + 1e-12f);  // alpha = rs / (p.Ap + eps)
  scal[2] = 0.0f;                          // reset rs_new accumulator
}

// x += alpha*p ; r -= alpha*Ap
__global__ __launch_bounds__(TPB) void kxr(float* __restrict__ x,
                                           float* __restrict__ r,
                                           const float* __restrict__ p,
                                           const float* __restrict__ w,
                                           const float* __restrict__ scal) {
  if (scal[3] != 0.0f) return;
  int i = blockIdx.x * TPB + threadIdx.x;
  if (i >= NF) return;
  float alpha = scal[4];
  x[i] = fmaf(alpha, p[i], x[i]);
  r[i] = fmaf(-alpha, w[i], r[i]);
}

__global__ void kbeta(float* __restrict__ scal) {
  if (scal[3] != 0.0f) return;
  float rs1 = scal[2];
  scal[5] = rs1 / (scal[0] + 1e-12f);                  // beta
  scal[0] = rs1;                                       // rs = rs1 (committed)
  scal[6] = (sqrtf(rs1) <= CG_TOL) ? 1.0f : 0.0f;      // done_next
}

// p = r + beta*p   (gated on OLD done, flag advances next iteration)
__global__ __launch_bounds__(TPB) void kp(float* __restrict__ p,
                                          const float* __restrict__ r,
                                          const float* __restrict__ scal) {
  if (scal[3] != 0.0f) return;
  int i = blockIdx.x * TPB + threadIdx.x;
  if (i >= NF) return;
  p[i] = fmaf(scal[5], p[i], r[i]);
}

extern "C" void kernel_launch(void* const* d_in, const int* in_sizes, int n_in,
                              void* d_out, int out_size, void* d_ws, size_t ws_size,
                              hipStream_t stream) {
  const float* u     = (const float*)d_in[0];
  const int*   eidx  = (const int*)d_in[1];       // [2, E]: row0=src, row1=dst
  const float* eattr = (const float*)d_in[2];     // [E, 2]
  const float* dtp   = (const float*)d_in[3];
  const float* gp    = (const float*)d_in[4];
  float* x = (float*)d_out;

  const int* src = eidx;
  const int* dst = eidx + N_EDGES;

  char* ws = (char*)d_ws;
  size_t off = 0;
  auto take = [&](size_t bytes) -> char* {
    char* q = ws + off;
    off += (bytes + 255) & ~(size_t)255;
    return q;
  };
  float* inv  = (float*)take(sizeof(float) * N_EDGES);
  float* rvec = (float*)take(sizeof(float) * NF);
  float* pvec = (float*)take(sizeof(float) * NF);
  float* wvec = (float*)take(sizeof(float) * NF);
  float* tmp  = (float*)take(sizeof(float) * NF);
  float* scal = (float*)take(sizeof(float) * 16);
  float* gate = scal + 3;

  // ---- setup: b, rhs = A^T b, r = p = rhs, rs = r.r ----
  kzero_setup<<<NB_NF, TPB, 0, stream>>>(tmp, x, scal);
  kprep<<<NB_E, TPB, 0, stream>>>(eattr, dst, inv, tmp /*slope_n*/);
  kb<<<NB_NF, TPB, 0, stream>>>(u, tmp, dtp, gp, wvec);          // w = b, tmp -> 0
  kd1t<<<NB_E, TPB, 0, stream>>>(src, dst, inv, u, wvec, tmp, gate);
  krhs<<<NB_NF, TPB, 0, stream>>>(wvec, tmp, dtp, rvec, pvec);
  kdot<<<NB_DOT, TPB, 0, stream>>>(rvec, rvec, scal + 2, gate, NF);
  kinit_scal<<<1, 1, 0, stream>>>(scal);

  // ---- 30 CG iterations on the normal equations ----
  for (int it = 0; it < CG_ITERS; ++it) {
    kzero_iter<<<NB_NF, TPB, 0, stream>>>(tmp, scal);            // gate <- done_next
    kd1<<<NB_E, TPB, 0, stream>>>(src, dst, inv, pvec, tmp, gate);
    kw<<<NB_NF, TPB, 0, stream>>>(pvec, u, tmp, dtp, wvec, gate); // w = A p, tmp -> 0
    kd1t<<<NB_E, TPB, 0, stream>>>(src, dst, inv, u, wvec, tmp, gate);
    kap<<<NB_NF, TPB, 0, stream>>>(wvec, tmp, dtp, gate);         // w = A^T A p
    kdot<<<NB_DOT, TPB, 0, stream>>>(pvec, wvec, scal + 1, gate, NF);
    kalpha<<<1, 1, 0, stream>>>(scal);
    kxr<<<NB_NF, TPB, 0, stream>>>(x, rvec, pvec, wvec, scal);
    kdot<<<NB_DOT, TPB, 0, stream>>>(rvec, rvec, scal + 2, gate, NF);
    kbeta<<<1, 1, 0, stream>>>(scal);
    kp<<<NB_NF, TPB, 0, stream>>>(pvec, rvec, scal);
  }
}